// PiPoolingLayer_57887569215662
// MI455X (gfx1250) — compile-verified
//
#include <hip/hip_runtime.h>

typedef __attribute__((ext_vector_type(2))) float v2f;
typedef __attribute__((ext_vector_type(8))) float v8f;

#define L_RES   64
#define N_ATOMS 14
#define M_TOT   896      // L_RES * N_ATOMS
#define D_FEAT  128
#define H_DIM   64
#define CUT2    25.0f    // 5.0^2

// ---------------------------------------------------------------------------
// Kernel 1: per-atom contact counts. rowcnt[v][i][a2] = #{l2 : mask[i, l2*14+a2]}
// mask[i,j] = (|p_i - p_j|^2 < 25) && (code[i] != code[j]); symmetric, so the
// column counts needed later are the same array.
// ---------------------------------------------------------------------------
__global__ void pp_count_kernel(const float* __restrict__ pos_wt,
                                const float* __restrict__ pos_mut,
                                const int*   __restrict__ ppi_wt,
                                const int*   __restrict__ ppi_mut,
                                float* __restrict__ R) {
  __shared__ float sx[M_TOT], sy[M_TOT], sz[M_TOT];
  __shared__ int   sc[M_TOT];
  const int variant = blockIdx.y;
  const float* __restrict__ pos = variant ? pos_mut : pos_wt;
  const int*   __restrict__ ppi = variant ? ppi_mut : ppi_wt;

  for (int k = threadIdx.x; k < M_TOT; k += blockDim.x) {
    sx[k] = pos[k * 3 + 0];
    sy[k] = pos[k * 3 + 1];
    sz[k] = pos[k * 3 + 2];
    sc[k] = ppi[k / N_ATOMS];
  }
  __syncthreads();

  const int i = blockIdx.x * blockDim.x + threadIdx.x;   // 0..895 exactly
  const float mx = sx[i], my = sy[i], mz = sz[i];
  const int mycode = sc[i];

  int cnt[N_ATOMS];
#pragma unroll
  for (int a2 = 0; a2 < N_ATOMS; ++a2) cnt[a2] = 0;

  for (int l2 = 0; l2 < L_RES; ++l2) {
    const int base = l2 * N_ATOMS;
#pragma unroll
    for (int a2 = 0; a2 < N_ATOMS; ++a2) {
      const int j = base + a2;
      const float dx = mx - sx[j];
      const float dy = my - sy[j];
      const float dz = mz - sz[j];
      const float d2 = dx * dx + dy * dy + dz * dz;
      cnt[a2] += ((d2 < CUT2) && (sc[j] != mycode)) ? 1 : 0;
    }
  }

  float* __restrict__ Ri = R + (size_t)variant * (M_TOT * N_ATOMS) + (size_t)i * N_ATOMS;
#pragma unroll
  for (int a2 = 0; a2 < N_ATOMS; ++a2) Ri[a2] = (float)cnt[a2];
}

// ---------------------------------------------------------------------------
// Kernel 2: U (128 x 16) row-major; col0 = u[d] = sum_h W1[h, d]   * W2[h]
//                                    col1 = v[d] = sum_h W1[h, D+d] * W2[h]
// remaining 14 columns zero (padding for the 16-wide WMMA B operand).
// ---------------------------------------------------------------------------
__global__ void pp_makeu_kernel(const float* __restrict__ W1,
                                const float* __restrict__ W2,
                                float* __restrict__ U) {
  const int t = threadIdx.x;                // 256 threads
  for (int k = t; k < D_FEAT * 16; k += 256) U[k] = 0.0f;
  __syncthreads();
  const int col = t >> 7;                   // 0 -> u, 1 -> v
  const int d   = t & 127;
  float acc = 0.0f;
  for (int h = 0; h < H_DIM; ++h)
    acc += W1[h * (2 * D_FEAT) + col * D_FEAT + d] * W2[h];
  U[d * 16 + col] = acc;
}

// ---------------------------------------------------------------------------
// Kernel 3: S = F @ U via V_WMMA_F32_16X16X4_F32 (fp32 matrix pipe).
// One wave per 16-row tile; 56 tiles x 2 variants = 112 waves (14 blocks x 8).
// A 16x4 layout (ISA 7.12.2): lane<16 holds K=0,1; lane>=16 holds K=2,3.
// B 4x16: VGPR r, lanes<16 -> K=r, lanes>=16 -> K=r+2; N = lane%16.
// C 16x16: VGPR r -> M = r + 8*(lane>=16), N = lane%16.
// Only N in {0,1} is useful; store S compact as [variant][row][2].
// ---------------------------------------------------------------------------
__global__ void pp_gemm_kernel(const float* __restrict__ feats_wt,
                               const float* __restrict__ feats_mut,
                               const float* __restrict__ U,
                               float* __restrict__ S) {
  const int gwave   = (blockIdx.x * blockDim.x + threadIdx.x) >> 5;  // 0..111
  const int lane    = threadIdx.x & 31;
  const int variant = gwave / 56;
  const int tile    = gwave % 56;
  const float* __restrict__ F = variant ? feats_mut : feats_wt;

  const int half = lane >> 4;       // 0 or 1
  const int r    = lane & 15;       // A: M row / B,C: N col
  const float* __restrict__ Arow = F + (size_t)(tile * 16 + r) * D_FEAT;

  v8f c = {};
  for (int kk = 0; kk < D_FEAT / 4; ++kk) {
    const int kb = kk * 4 + half * 2;
    v2f a, b;
    a.x = Arow[kb];
    a.y = Arow[kb + 1];
    b.x = U[(kb + 0) * 16 + r];
    b.y = U[(kb + 1) * 16 + r];
    c = __builtin_amdgcn_wmma_f32_16x16x4_f32(
        /*neg_a=*/false, a, /*neg_b=*/false, b,
        /*c_mod=*/(short)0, c, /*reuse_a=*/false, /*reuse_b=*/false);
  }

  if (r < 2) {                      // lanes 0,1,16,17 hold the two useful cols
    const int rowbase = tile * 16 + half * 8;
    float* __restrict__ Sv = S + (size_t)variant * (M_TOT * 2);
#pragma unroll
    for (int rr = 0; rr < 8; ++rr)
      Sv[(rowbase + rr) * 2 + r] = c[rr];
  }
}

// ---------------------------------------------------------------------------
// Kernel 4: 196 outputs.
// out[a1,a2] = [mut] - [wt] + b2, with per-variant term:
//   sum_l s_u[l*14+a1]*R[l*14+a1][a2] + sum_l s_v[l*14+a2]*R[l*14+a2][a1]
//   + (b1.w2) * sum_l R[l*14+a1][a2]
// ---------------------------------------------------------------------------
__global__ void pp_final_kernel(const float* __restrict__ R,
                                const float* __restrict__ S,
                                const float* __restrict__ b1,
                                const float* __restrict__ W2,
                                const float* __restrict__ b2,
                                float* __restrict__ out) {
  const int idx = threadIdx.x;
  if (idx >= N_ATOMS * N_ATOMS) return;
  const int a1 = idx / N_ATOMS;
  const int a2 = idx % N_ATOMS;

  float b1w2 = 0.0f;
  for (int h = 0; h < H_DIM; ++h) b1w2 += b1[h] * W2[h];

  float res = b2[0];
  for (int variant = 0; variant < 2; ++variant) {
    const float* __restrict__ Rv = R + (size_t)variant * (M_TOT * N_ATOMS);
    const float* __restrict__ Sv = S + (size_t)variant * (M_TOT * 2);
    float t = 0.0f, cnt = 0.0f;
    for (int l = 0; l < L_RES; ++l) {
      const int i = l * N_ATOMS + a1;
      const int j = l * N_ATOMS + a2;
      const float ria2 = Rv[i * N_ATOMS + a2];
      t   += Sv[i * 2 + 0] * ria2;
      t   += Sv[j * 2 + 1] * Rv[j * N_ATOMS + a1];
      cnt += ria2;
    }
    t += b1w2 * cnt;
    res += variant ? t : -t;
  }
  out[idx] = res;
}

// ---------------------------------------------------------------------------
extern "C" void kernel_launch(void* const* d_in, const int* in_sizes, int n_in,
                              void* d_out, int out_size, void* d_ws, size_t ws_size,
                              hipStream_t stream) {
  (void)in_sizes; (void)n_in; (void)out_size; (void)ws_size;

  const float* feats_wt  = (const float*)d_in[0];
  const float* feats_mut = (const float*)d_in[1];
  const float* pos_wt    = (const float*)d_in[2];
  const float* pos_mut   = (const float*)d_in[3];
  const int*   ppi_wt    = (const int*)d_in[4];
  const int*   ppi_mut   = (const int*)d_in[5];
  const float* W1        = (const float*)d_in[6];
  const float* b1        = (const float*)d_in[7];
  const float* W2        = (const float*)d_in[8];
  const float* b2        = (const float*)d_in[9];
  float* out = (float*)d_out;

  float* R = (float*)d_ws;                    // 2 * 896 * 14 = 25088 floats
  float* U = R + 2 * M_TOT * N_ATOMS;         // 128 * 16     =  2048 floats
  float* S = U + D_FEAT * 16;                 // 2 * 896 * 2  =  3584 floats

  pp_count_kernel<<<dim3(7, 2), 128, 0, stream>>>(pos_wt, pos_mut, ppi_wt, ppi_mut, R);
  pp_makeu_kernel<<<1, 256, 0, stream>>>(W1, W2, U);
  pp_gemm_kernel<<<14, 256, 0, stream>>>(feats_wt, feats_mut, U, S);
  pp_final_kernel<<<1, 224, 0, stream>>>(R, S, b1, W2, b2, out);
}